// HyenaFilter_54520314856278
// MI455X (gfx1250) — compile-verified
//
#include <hip/hip_runtime.h>
#include <hip/hip_bf16.h>

typedef __attribute__((ext_vector_type(2))) float v2f;
typedef __attribute__((ext_vector_type(8))) float v8f;

#define B_ 4
#define S_ 8192
#define D_ 1024
#define O_ 4

// ---------------------------------------------------------------------------
// Kernel 1: proj[(s*B + b)*O + o] = sum_d x[b][s][d] * w[o][d]
// One wave32 per 16-row tile of the flattened (b,s) dimension.
// V_WMMA_F32_16X16X4_F32, K stepped by 4 across D=1024 (256 WMMAs/tile).
// B-fragment lanes m>=O read a clamped valid row and are zeroed with a
// branch-free select (v_cndmask), keeping the inner loop divergence-free.
// ---------------------------------------------------------------------------
__global__ void __launch_bounds__(256)
hyena_proj_wmma(const float* __restrict__ x,
                const float* __restrict__ w,
                float* __restrict__ proj)
{
    const int  wave = (int)((blockIdx.x * blockDim.x + threadIdx.x) >> 5);
    const int  lane = (int)(threadIdx.x & 31);
    const int  m    = lane & 15;   // N column for B/D, M row for A
    const int  kh   = lane >> 4;   // K half: lanes 16-31 hold K=2,3
    const long row0 = (long)wave * 16;

    // A fragment: lane holds x[row0+m][k + 2*kh], x[row0+m][k + 2*kh + 1]
    const float* xrow = x + (row0 + m) * D_ + 2 * kh;
    // B fragment: lane m holds w[m][k + 2*kh .. +1] for m<O, else 0.
    // Clamp the row so every lane loads a valid address; zero via select.
    const float* wrow  = w + (long)(m & (O_ - 1)) * D_ + 2 * kh;
    const bool   bvalid = (m < O_);

    v8f acc = {};
    #pragma unroll 4
    for (int k = 0; k < D_; k += 4) {
        v2f a;
        a.x = __builtin_nontemporal_load(xrow + k);       // x streamed once: NT
        a.y = __builtin_nontemporal_load(xrow + k + 1);
        float w0 = wrow[k];                               // 16 KB, cache-hot
        float w1 = wrow[k + 1];
        v2f bm;
        bm.x = bvalid ? w0 : 0.0f;                        // v_cndmask, no branch
        bm.y = bvalid ? w1 : 0.0f;
        // D = A(16x4) * B(4x16) + C ; emits v_wmma_f32_16x16x4_f32
        acc = __builtin_amdgcn_wmma_f32_16x16x4_f32(false, a, false, bm,
                                                    (short)0, acc, false, false);
    }

    // D layout: VGPR j -> M = j + 8*kh, N = m. Only N < O_ columns are real.
    if (bvalid) {
        #pragma unroll
        for (int j = 0; j < 8; ++j) {
            long r = row0 + j + 8 * kh;       // flattened (b,s) row
            int  b = (int)(r / S_);
            int  s = (int)(r % S_);
            proj[((long)s * B_ + b) * O_ + m] = acc[j];
        }
    }
}

// ---------------------------------------------------------------------------
// Kernel 2: chunked linear scan.
//   y_t[o] = c[o,d] * y_{t-1}[o] + proj[t,b,o];  out[b,t,d] = sum_o y_t[o]
// All decay factors are exp(v) with v <= -0.1, so c^2048 <= e^-204.8 == 0 in
// f32. Each 1024-step chunk re-warms from zero over the preceding <=2048
// steps, giving 8x scan parallelism with f32-exact results. Per step:
// 1 broadcast float4 load (WGP$/L2 hit), 4 FMA, 3 ADD, 1 NT coalesced store.
// ---------------------------------------------------------------------------
#define CHUNK 1024
#define WARM  2048

__global__ void __launch_bounds__(64)
hyena_recurrence(const float* __restrict__ proj,
                 const float* __restrict__ fc,
                 float* __restrict__ out)
{
    const int d     = (int)(blockIdx.x * 64 + threadIdx.x);
    const int chunk = (int)blockIdx.y;
    const int b     = (int)blockIdx.z;

    const float c0 = __expf(fc[0 * D_ + d]);
    const float c1 = __expf(fc[1 * D_ + d]);
    const float c2 = __expf(fc[2 * D_ + d]);
    const float c3 = __expf(fc[3 * D_ + d]);

    const int t0 = chunk * CHUNK;
    int t = t0 - WARM;
    if (t < 0) t = 0;

    const float4* p4 = (const float4*)proj;
    float y0 = 0.f, y1 = 0.f, y2 = 0.f, y3 = 0.f;

    // warm-up: build state, no stores (history beyond 2048 steps underflows)
    for (; t < t0; ++t) {
        float4 p = p4[(long)t * B_ + b];
        y0 = fmaf(c0, y0, p.x);
        y1 = fmaf(c1, y1, p.y);
        y2 = fmaf(c2, y2, p.z);
        y3 = fmaf(c3, y3, p.w);
    }

    float* o = out + (long)b * S_ * D_ + d;
    __builtin_prefetch(&p4[(long)t0 * B_ + b], 0, 1);

    #pragma unroll 4
    for (; t < t0 + CHUNK; ++t) {
        float4 p = p4[(long)t * B_ + b];
        y0 = fmaf(c0, y0, p.x);
        y1 = fmaf(c1, y1, p.y);
        y2 = fmaf(c2, y2, p.z);
        y3 = fmaf(c3, y3, p.w);
        // write-once 128 MB stream: nontemporal, coalesced over d
        __builtin_nontemporal_store((y0 + y1) + (y2 + y3), o + (long)t * D_);
    }
}

// ---------------------------------------------------------------------------
extern "C" void kernel_launch(void* const* d_in, const int* in_sizes, int n_in,
                              void* d_out, int out_size, void* d_ws, size_t ws_size,
                              hipStream_t stream) {
    const float* x  = (const float*)d_in[0];   // (B,S,D) f32
    const float* w  = (const float*)d_in[1];   // (O,D)   f32
    const float* fc = (const float*)d_in[2];   // (O,D)   f32
    float*       out  = (float*)d_out;         // (B,S,D) f32
    float*       proj = (float*)d_ws;          // (S,B,O) f32 scratch: 512 KB

    // GEMM: 2048 tile-waves, 8 waves per 256-thread block -> 256 blocks
    const int nwaves = (B_ * S_) / 16;
    hyena_proj_wmma<<<nwaves / 8, 256, 0, stream>>>(x, w, proj);

    // Scan: 16 d-groups x 8 chunks x 4 batches = 512 blocks of 2 waves
    dim3 grid(D_ / 64, S_ / CHUNK, B_);
    hyena_recurrence<<<grid, 64, 0, stream>>>(proj, fc, out);
}